// NeoLoss_48223892800204
// MI455X (gfx1250) — compile-verified
//
#include <hip/hip_runtime.h>
#include <hip/hip_bf16.h>

// BCE loss (reduce=False) with torch-style -100 log clamp and "already
// correct -> 0" masking. Memory-bound streaming kernel:
//   12 B/elem * 16.7M elem ~= 201 MB  ->  ~8.6 us at 23.3 TB/s.
// Strategy: wave32-friendly 256-thread blocks, b128 vector memory ops,
// non-temporal (TH=NT) hints since every byte is touched exactly once and
// the working set exceeds the 192 MB L2. One v_log_f32 per element.

// clang ext_vector type: required for __builtin_nontemporal_load/store
// (HIP's float4 is a class and is rejected by the builtin).
typedef float v4f __attribute__((ext_vector_type(4)));

#define LN2F 0.6931471805599453f

__device__ __forceinline__ float bce_elem(float p, float y) {
    // y is binary (0.0f or 1.0f). Probability assigned to the true class:
    float t = (y != 0.0f) ? p : (1.0f - p);
    // -ln(t) = -log2(t) * ln(2); clamp matches max(log, -100) in reference.
    float nll = fminf(-__log2f(t) * LN2F, 100.0f);
    // zero out when prediction already on the correct side of 0.5
    bool correct = (p - 0.5f) * (2.0f * y - 1.0f) > 0.0f;
    return correct ? 0.0f : nll;
}

__global__ __launch_bounds__(256) void
NeoLoss_48223892800204_kernel(const v4f* __restrict__ p4,
                              const v4f* __restrict__ y4,
                              v4f* __restrict__ o4,
                              const float* __restrict__ p1,
                              const float* __restrict__ y1,
                              float* __restrict__ o1,
                              int n4, int n) {
    int gid = blockIdx.x * blockDim.x + threadIdx.x;

    if (gid < n4) {
        // global_load_b128 th:TH_NT — streaming, single-touch data
        v4f pv = __builtin_nontemporal_load(&p4[gid]);
        v4f yv = __builtin_nontemporal_load(&y4[gid]);
        v4f ov;
        ov.x = bce_elem(pv.x, yv.x);
        ov.y = bce_elem(pv.y, yv.y);
        ov.z = bce_elem(pv.z, yv.z);
        ov.w = bce_elem(pv.w, yv.w);
        // global_store_b128 th:TH_NT
        __builtin_nontemporal_store(ov, &o4[gid]);
    }

    // scalar tail (empty for N = 16,777,216, kept for generality)
    int tail_base = n4 * 4;
    int tail_idx = tail_base + gid;
    if (gid < (n - tail_base)) {
        float p = __builtin_nontemporal_load(&p1[tail_idx]);
        float y = __builtin_nontemporal_load(&y1[tail_idx]);
        __builtin_nontemporal_store(bce_elem(p, y), &o1[tail_idx]);
    }
}

extern "C" void kernel_launch(void* const* d_in, const int* in_sizes, int n_in,
                              void* d_out, int out_size, void* d_ws, size_t ws_size,
                              hipStream_t stream) {
    const float* y_pred   = (const float*)d_in[0];
    const float* y_actual = (const float*)d_in[1];
    float* out            = (float*)d_out;

    int n  = in_sizes[0];
    int n4 = n >> 2;

    const int block = 256;                       // 8 wave32 waves per block
    int work = (n4 > 0) ? n4 : (n - n4 * 4);     // threads needed
    int grid = (work + block - 1) / block;
    if (grid < 1) grid = 1;

    NeoLoss_48223892800204_kernel<<<grid, block, 0, stream>>>(
        (const v4f*)y_pred, (const v4f*)y_actual, (v4f*)out,
        y_pred, y_actual, out, n4, n);
}